// HadamardTransform_40243843564261
// MI455X (gfx1250) — compile-verified
//
#include <hip/hip_runtime.h>

typedef __attribute__((ext_vector_type(2))) float v2f;
typedef __attribute__((ext_vector_type(8))) float v8f;
typedef unsigned int u32;
typedef __attribute__((ext_vector_type(4))) u32 v4u;
typedef __attribute__((ext_vector_type(8))) int v8i;
typedef __attribute__((ext_vector_type(4))) int v4i;

#define GROUP_PAD 20  // 16 data floats + 4 pad floats per group: 80B, 16B-aligned, conflict-free strides

__device__ __forceinline__ int lidx(int i) {
    return (i >> 4) * GROUP_PAD + (i & 15);
}

// Sylvester Hadamard sign: H[r][c] = (-1)^popcount(r & c), via sign-bit xor
__device__ __forceinline__ float hsgn(int r, int c) {
    u32 bits = 0x3F800000u ^ ((u32)(__popc(r & c) & 1) << 31);
    return __builtin_bit_cast(float, bits);
}

// One 16-point Hadamard pass over this wave's 16x16 tile, in place in LDS.
// B element X[K][N] lives at lds[base + K*strideK] (base already includes N = lane&15).
// D element (M, N) is written back to lds[base + M*strideK].
// A is the 16x16 H matrix (chunked 16x4), generated in registers.
__device__ __forceinline__ void had16_pass(float* lds, int base, int strideK,
                                           int m, int half, int kb) {
    v8f c = {};
#pragma unroll
    for (int k = 0; k < 4; ++k) {
        const int K0 = 4 * k + kb;      // lane-half selects K pair within the 16x4 chunk
        v2f a, b;
        a.x = hsgn(m, K0);
        a.y = hsgn(m, K0 + 1);
        b.x = lds[base + K0 * strideK];
        b.y = lds[base + (K0 + 1) * strideK];
        c = __builtin_amdgcn_wmma_f32_16x16x4_f32(
                /*neg_a=*/false, a, /*neg_b=*/false, b,
                /*c_mod=*/(short)0, c, /*reuse_a=*/false, /*reuse_b=*/false);
    }
    // C/D layout: vgpr r -> row M = r + 8*half, col N = lane&15 (already in base)
#pragma unroll
    for (int r = 0; r < 8; ++r) {
        lds[base + (r + 8 * half) * strideK] = c[r];
    }
}

__launch_bounds__(512)
__global__ void fwht4096_kernel(const float* __restrict__ x, float* __restrict__ y) {
    __shared__ __align__(16) float lds[256 * GROUP_PAD];  // 20 KB

    const int row  = blockIdx.x;
    const float* __restrict__ xr = x + (size_t)row * 4096;
    float* __restrict__ yr       = y + (size_t)row * 4096;

    const int tid  = threadIdx.x;
    const int lane = tid & 31;
    const int wave = tid >> 5;     // tile id 0..15
    const int n    = lane & 15;    // N (B/D column) == M (A row) for this lane
    const int half = lane >> 4;    // selects K pair (A/B) and +8 row offset (D)
    const int kb   = half << 1;

    // ---- stage in via Tensor Data Mover: one 16KB 1-D descriptor per block.
    // TDM pad feature (pad_interval=16 DWORDs, pad_amount=4 DWORDs) reproduces the
    // 16->20-float padded LDS group layout in hardware.
    if (wave == 0) {
        const u32 lds_base = (u32)(size_t)(void*)&lds[0];          // WG-relative LDS byte offset
        const unsigned long long ga = (unsigned long long)(size_t)xr;
        v4u g0;
        g0.x = 1u;                                                 // count=1, is_restore=0, gather=0
        g0.y = lds_base;                                           // lds_addr
        g0.z = (u32)ga;                                            // global_addr[31:0]
        g0.w = ((u32)(ga >> 32) & 0x01FFFFFFu) | 0x80000000u;      // global_addr[56:32] | type=2
        v8i g1;
        g1[0] = 0x06D20000;        // wg_mask=0 | data_size=2(4B) | pad_enable | pad_interval=3 | pad_amount=3
        g1[1] = (int)(4096u << 16);// tensor_dim0[15:0]=4096 at bits[63:48]
        g1[2] = (int)(1u << 16);   // tensor_dim0[31:16]=0, tensor_dim1=1
        g1[3] = (int)(4096u << 16);// tensor_dim1[31:16]=0, tile_dim0=4096 at bits[127:112]
        g1[4] = 0;                 // tile_dim1=0 (unused), tile_dim2=0
        g1[5] = 4096;              // tensor_dim0_stride[31:0]
        g1[6] = 0;                 // stride hi, tensor_dim1_stride lo
        g1[7] = 0;
        const v4i gz = {0, 0, 0, 0};
        const v8i gz8 = {0, 0, 0, 0, 0, 0, 0, 0};
        __builtin_amdgcn_tensor_load_to_lds(g0, g1, gz, gz, gz8, 0);
        __builtin_amdgcn_s_wait_tensorcnt(0);
    }
    __syncthreads();

    // index i = i2*256 + i1*16 + i0, digits in 0..15; LDS addr = (i2*16+i1)*20 + i0

    // pass 0: transform along i0 (K stride 1); tile t covers groups [16t, 16t+16)
    had16_pass(lds, (16 * wave + n) * GROUP_PAD, 1, n, half, kb);
    __syncthreads();

    // pass 1: transform along i1 (K stride 20); tile t = i2, N = i0
    had16_pass(lds, wave * 16 * GROUP_PAD + n, GROUP_PAD, n, half, kb);
    __syncthreads();

    // pass 2: transform along i2 (K stride 320); tile t = i1, N = i0
    had16_pass(lds, wave * GROUP_PAD + n, 16 * GROUP_PAD, n, half, kb);
    __syncthreads();

    // ---- stage out with exact power-of-two normalization 1/sqrt(4096) ----
    const float s = 1.0f / 64.0f;
#pragma unroll
    for (int j = 0; j < 2; ++j) {
        const int idx4 = tid + j * 512;
        float4 v = *((const float4*)&lds[lidx(idx4 * 4)]);
        v.x *= s; v.y *= s; v.z *= s; v.w *= s;
        ((float4*)yr)[idx4] = v;
    }
}

extern "C" void kernel_launch(void* const* d_in, const int* in_sizes, int n_in,
                              void* d_out, int out_size, void* d_ws, size_t ws_size,
                              hipStream_t stream) {
    const float* x = (const float*)d_in[0];
    float* y = (float*)d_out;
    const int total = in_sizes[0];        // 4 * 2048 * 4096
    const int rows  = total / 4096;       // 8192 rows of length 4096
    fwht4096_kernel<<<rows, 512, 0, stream>>>(x, y);
}